// RNN_41506563948968
// MI455X (gfx1250) — compile-verified
//
#include <hip/hip_runtime.h>

#define DH 2048
#define DI 64
#define DO 64
#define T_STEPS 512
#define BATCH 128

typedef __attribute__((ext_vector_type(2))) float v2f;
typedef __attribute__((ext_vector_type(8))) float v8f;

__device__ __forceinline__ float fast_sigmoid(float x) {
    return 1.0f / (1.0f + __expf(-x));
}

// ---------------------------------------------------------------------------
// Kernel 0: sigr0 = sigmoid(r0)   (2048 x 128 elementwise)
// ---------------------------------------------------------------------------
__global__ void k_sigr0(const float* __restrict__ r0, float* __restrict__ sigr0) {
    int i = blockIdx.x * blockDim.x + threadIdx.x;
    if (i < DH * BATCH) sigr0[i] = fast_sigmoid(r0[i]);
}

// ---------------------------------------------------------------------------
// Kernel 1: base = (1-alpha)*r0 + alpha * (|_w| @ sigmoid(r0))
// One wave per 16x16 tile of (2048 x 128). 8 waves / block -> 128 blocks.
// K-loop over DH with V_WMMA_F32_16X16X4_F32.
// ---------------------------------------------------------------------------
__global__ __launch_bounds__(256) void k_base(
    const float* __restrict__ w_, const float* __restrict__ sigr0,
    const float* __restrict__ r0, const float* __restrict__ taus,
    float* __restrict__ base)
{
    const int lane = threadIdx.x & 31;
    const int wav  = threadIdx.x >> 5;
    const int tile = blockIdx.x * 8 + wav;   // 0..1023
    const int h0 = (tile >> 3) * 16;         // row tile (DH/16 = 128)
    const int b0 = (tile & 7) * 16;          // col tile (BATCH/16 = 8)

    const int col   = lane & 15;             // M for A, N for B
    const int khalf = (lane >> 4) << 1;      // K offset 0 or 2
    const int mhi   = (lane >> 4) << 3;      // C/D row offset 0 or 8

    v8f acc = {};
    const float* wrow = w_ + (size_t)(h0 + col) * DH;
    for (int k0 = 0; k0 < DH; k0 += 4) {
        const int ka = k0 + khalf;
        v2f a = *(const v2f*)(wrow + ka);    // _w[row, ka..ka+1]
        a.x = fabsf(a.x);
        a.y = fabsf(a.y);
        v2f b;
        b.x = sigr0[(size_t)ka * BATCH + b0 + col];
        b.y = sigr0[(size_t)(ka + 1) * BATCH + b0 + col];
        acc = __builtin_amdgcn_wmma_f32_16x16x4_f32(false, a, false, b,
                                                    (short)0, acc, false, false);
    }

#pragma unroll
    for (int r = 0; r < 8; ++r) {
        const int h = h0 + r + mhi;
        const float alpha = 1.0f / taus[h];          // DT = 1.0
        const size_t idx = (size_t)h * BATCH + b0 + col;
        base[idx] = (1.0f - alpha) * r0[idx] + alpha * acc[r];
    }
}

// ---------------------------------------------------------------------------
// Kernel 2: per (t, b-tile):
//   xs[t,h,b]  = base[h,b] + (win @ u[t])[h,b] + noise[t,h,b]
//   out[t,o,b] = wout @ sigmoid(xs[t]) + bias
// grid = T * (BATCH/16) blocks of 256 threads (8 waves).
// ---------------------------------------------------------------------------
__global__ __launch_bounds__(256) void k_main(
    const float* __restrict__ u,    const float* __restrict__ noise,
    const float* __restrict__ win,  const float* __restrict__ wout,
    const float* __restrict__ bias, const float* __restrict__ base,
    float* __restrict__ out, float* __restrict__ xs)
{
    __shared__ float su[16][DI];          // u tile, transposed: [n][d]
    __shared__ float sstage[8][16][16];   // per-wave sigmoid tile: [wave][n][m]
    __shared__ float sout[8][DO][16];     // per-wave output partials

    const int tid  = threadIdx.x;
    const int lane = tid & 31;
    const int wav  = tid >> 5;
    const int t  = blockIdx.x >> 3;
    const int b0 = (blockIdx.x & 7) * 16;

    // cooperative load of u[t, :, b0:b0+16] (transposed into LDS)
    {
        const float* ut = u + (size_t)t * DI * BATCH;
#pragma unroll
        for (int i = 0; i < 4; ++i) {
            const int e = tid + i * 256;         // 0..1023
            const int n = e & 15, d = e >> 4;
            su[n][d] = ut[(size_t)d * BATCH + b0 + n];
        }
    }
    __syncthreads();

    const int col   = lane & 15;
    const int khalf = (lane >> 4) << 1;
    const int mhi   = (lane >> 4) << 3;

    v8f oacc[4] = {{}, {}, {}, {}};       // DO=64 -> 4 row tiles of 16

    for (int ht = wav; ht < DH / 16; ht += 8) {
        const int h0 = ht * 16;

        // ---- inp tile: win[h0:h0+16, 0:64] @ u-tile (K = 64, 16 WMMAs) ----
        v8f acc = {};
        const float* wrow = win + (size_t)(h0 + col) * DI;
#pragma unroll
        for (int kk = 0; kk < 16; ++kk) {
            const int ka = kk * 4 + khalf;
            v2f a = *(const v2f*)(wrow + ka);        // A[M=col, K=ka..ka+1]
            v2f b = *(const v2f*)(&su[col][ka]);     // B[K=ka..ka+1, N=col]
            acc = __builtin_amdgcn_wmma_f32_16x16x4_f32(false, a, false, b,
                                                        (short)0, acc, false, false);
        }

        // ---- xs = acc + base + noise ; stage sigmoid(xs) in LDS ----
        const size_t nb = ((size_t)t * DH + h0) * BATCH + b0 + col;
#pragma unroll
        for (int r = 0; r < 8; ++r) {
            const int m = r + mhi;
            const float x = acc[r]
                          + base[(size_t)(h0 + m) * BATCH + b0 + col]
                          + noise[nb + (size_t)m * BATCH];
            xs[nb + (size_t)m * BATCH] = x;
            sstage[wav][col][m] = fast_sigmoid(x);   // transposed: [n][m]
        }

        // ---- out tiles: oacc[ot] += wout[ot*16:.., h0:h0+16] @ s (K = 16) ----
#pragma unroll
        for (int kk = 0; kk < 4; ++kk) {
            const int kl = kk * 4 + khalf;
            v2f b = *(const v2f*)(&sstage[wav][col][kl]);  // B[K=kl..kl+1, N=col]
#pragma unroll
            for (int ot = 0; ot < 4; ++ot) {
                const float* wo = wout + (size_t)(ot * 16 + col) * DH + h0;
                v2f a = *(const v2f*)(wo + kl);            // A[M=col, K=kl..kl+1]
                oacc[ot] = __builtin_amdgcn_wmma_f32_16x16x4_f32(false, a, false, b,
                                                                 (short)0, oacc[ot],
                                                                 false, false);
            }
        }
    }

    // ---- cross-wave reduction of output partials ----
#pragma unroll
    for (int ot = 0; ot < 4; ++ot)
#pragma unroll
        for (int r = 0; r < 8; ++r)
            sout[wav][ot * 16 + r + mhi][col] = oacc[ot][r];
    __syncthreads();

#pragma unroll
    for (int i = 0; i < 4; ++i) {
        const int e = tid + i * 256;          // 0..1023 -> DO x 16
        const int o = e >> 4, n = e & 15;
        float s = 0.0f;
#pragma unroll
        for (int wv = 0; wv < 8; ++wv) s += sout[wv][o][n];
        out[((size_t)t * DO + o) * BATCH + b0 + n] = s + bias[o];
    }
}

// ---------------------------------------------------------------------------
extern "C" void kernel_launch(void* const* d_in, const int* in_sizes, int n_in,
                              void* d_out, int out_size, void* d_ws, size_t ws_size,
                              hipStream_t stream) {
    const float* u     = (const float*)d_in[0];
    const float* r0    = (const float*)d_in[1];
    const float* noise = (const float*)d_in[2];
    const float* win   = (const float*)d_in[3];
    const float* w_    = (const float*)d_in[4];
    // d_in[5] = m_diag: unused (|_w * (+-1)| == |_w|)
    const float* wout  = (const float*)d_in[6];
    const float* bias  = (const float*)d_in[7];
    const float* taus  = (const float*)d_in[8];

    float* out = (float*)d_out;                              // (T, DO, B)
    float* xs  = out + (size_t)T_STEPS * DO * BATCH;         // (T, DH, B)

    float* sigr0 = (float*)d_ws;                             // DH*BATCH f32
    float* base  = sigr0 + (size_t)DH * BATCH;               // DH*BATCH f32

    k_sigr0<<<(DH * BATCH + 255) / 256, 256, 0, stream>>>(r0, sigr0);
    k_base<<<(DH / 16) * (BATCH / 16) / 8, 256, 0, stream>>>(w_, sigr0, r0, taus, base);
    k_main<<<T_STEPS * (BATCH / 16), 256, 0, stream>>>(u, noise, win, wout, bias,
                                                       base, out, xs);
}